// GCN_3908420240152
// MI455X (gfx1250) — compile-verified
//
#include <hip/hip_runtime.h>
#include <hip/hip_bf16.h>

#define NDIM 256
#define EPS 1e-5f

typedef __attribute__((ext_vector_type(2))) float v2f;
typedef __attribute__((ext_vector_type(8))) float v8f;

// ---------------- utility kernels ----------------

__global__ void zero_kernel(float* __restrict__ p, long long n) {
    long long stride = (long long)gridDim.x * blockDim.x;
    for (long long i = (long long)blockIdx.x * blockDim.x + threadIdx.x; i < n; i += stride)
        p[i] = 0.0f;
}

__global__ void deg_kernel(const int* __restrict__ src, const int* __restrict__ dst,
                           float* __restrict__ outdeg, float* __restrict__ indeg, int nE) {
    int i = blockIdx.x * blockDim.x + threadIdx.x;
    if (i >= nE) return;
    atomicAdd(&outdeg[src[i]], 1.0f);
    atomicAdd(&indeg[dst[i]], 1.0f);
}

__global__ void cnorm_kernel(const float* __restrict__ outdeg, const float* __restrict__ indeg,
                             float* __restrict__ c_src, float* __restrict__ c_dst, int n) {
    int i = blockIdx.x * blockDim.x + threadIdx.x;
    if (i >= n) return;
    c_src[i] = rsqrtf(fmaxf(outdeg[i], 1.0f));
    c_dst[i] = rsqrtf(fmaxf(indeg[i], 1.0f));
}

// xs[i][:] = x[i][:] * c[i]   (float4 vectorized; 64 float4 per row)
__global__ void scale_rows_kernel(const float* __restrict__ x, const float* __restrict__ c,
                                  float* __restrict__ xs, int nRows) {
    long long gid = (long long)blockIdx.x * blockDim.x + threadIdx.x;
    long long total = (long long)nRows * (NDIM / 4);
    if (gid >= total) return;
    int row = (int)(gid >> 6);
    float cv = c[row];
    float4 v = ((const float4*)x)[gid];
    v.x *= cv; v.y *= cv; v.z *= cv; v.w *= cv;
    ((float4*)xs)[gid] = v;
}

// Edge-parallel scatter-add: 64 threads per edge, 4 floats each (float4 gather, f32 atomics).
// Gathers/accumulates stay resident in the 192MB L2 (node matrix = 51.2MB).
__global__ void scatter_kernel(const float* __restrict__ xs, const int* __restrict__ src,
                               const int* __restrict__ dst, float* __restrict__ agg, int nE) {
    long long gid = (long long)blockIdx.x * blockDim.x + threadIdx.x;
    int e = (int)(gid >> 6);
    if (e >= nE) return;
    int c4 = (int)(gid & 63);
    int s = src[e];
    int d = dst[e];
    float4 v = ((const float4*)(xs + (long long)s * NDIM))[c4];
    float* p = agg + (long long)d * NDIM + c4 * 4;
    atomicAdd(p + 0, v.x);
    atomicAdd(p + 1, v.y);
    atomicAdd(p + 2, v.z);
    atomicAdd(p + 3, v.w);
}

// ---------------- WMMA f32 GEMM:  C = (A * c_dst[row]) @ W + bias ----------------
// Block = 8 waves; each wave owns a 16x16 output tile; block covers 128 rows x 16 cols.
// W column-panel (256x16 = 16KB) staged in LDS; A fragments are v2f global loads.
__global__ __launch_bounds__(256) void wmma_gemm_kernel(
    const float* __restrict__ A, const float* __restrict__ cdst,
    const float* __restrict__ W, const float* __restrict__ bias,
    float* __restrict__ C, int nRows) {
    __shared__ float wtile[NDIM * 16];  // wtile[k*16 + n] = W[k][col0+n]

    const int col0 = blockIdx.y * 16;
    const int tid = threadIdx.x;

    // cooperative stage of the W panel (coalesced 64B groups)
    for (int g = tid; g < NDIM * 16; g += 256) {
        int k = g >> 4;
        int n = g & 15;
        wtile[g] = W[k * NDIM + col0 + n];
    }
    __syncthreads();

    const int wave = tid >> 5;
    const int lane = tid & 31;
    const int row0 = blockIdx.x * 128 + wave * 16;
    if (row0 >= nRows) return;  // whole-wave guard: EXEC all-1s for WMMA

    const int m = row0 + (lane & 15);       // A row / C row base for this lane
    const int n = lane & 15;                // C column within tile
    const int khalf = (lane >> 4) << 1;     // 0 for lanes 0-15, 2 for lanes 16-31
    const float cd = cdst[m];
    const float* __restrict__ arow = A + (long long)m * NDIM;

    v8f acc = {};
#pragma unroll 8
    for (int k = 0; k < NDIM; k += 4) {
        const int kk = k + khalf;
        // A frag: lane holds A[m][kk], A[m][kk+1] (8B-aligned contiguous pair), fused c_dst scale
        v2f a = *(const v2f*)(arow + kk) * cd;
        // B frag: lane holds W[kk][col0+n], W[kk+1][col0+n] from LDS (bank-conflict free)
        v2f b;
        b.x = wtile[kk * 16 + n];
        b.y = wtile[(kk + 1) * 16 + n];
        acc = __builtin_amdgcn_wmma_f32_16x16x4_f32(
            /*neg_a=*/false, a, /*neg_b=*/false, b,
            /*c_mod=*/(short)0, acc, /*reuse_a=*/false, /*reuse_b=*/false);
    }

    const float bv = bias[col0 + n];
    const int rbase = row0 + ((lane >> 4) << 3);  // lanes 16-31 hold M rows 8..15
    float* __restrict__ cptr = C + (long long)rbase * NDIM + col0 + n;
#pragma unroll
    for (int r = 0; r < 8; ++r)
        cptr[(long long)r * NDIM] = acc[r] + bv;
}

// ---------------- BatchNorm ----------------

// column-wise partial sums; thread t owns column t (coalesced row reads)
__global__ void bn_stats_kernel(const float* __restrict__ h, float* __restrict__ sum,
                                float* __restrict__ sumsq, int nRows) {
    int t = threadIdx.x;
    float s = 0.0f, s2 = 0.0f;
    for (int i = blockIdx.x; i < nRows; i += gridDim.x) {
        float v = h[(long long)i * NDIM + t];
        s += v;
        s2 += v * v;
    }
    atomicAdd(&sum[t], s);
    atomicAdd(&sumsq[t], s2);
}

__global__ void bn_finalize_kernel(const float* __restrict__ sum, const float* __restrict__ sumsq,
                                   const float* __restrict__ g, const float* __restrict__ be,
                                   float* __restrict__ scl, float* __restrict__ shf, float invN) {
    int t = threadIdx.x;
    float mu = sum[t] * invN;
    float var = sumsq[t] * invN - mu * mu;
    float s = g[t] * rsqrtf(var + EPS);
    scl[t] = s;
    shf[t] = be[t] - mu * s;
}

// xs[i][j] = relu(h[i][j]*scl[j] + shf[j]) * c_src[i]   (produces next layer's gather input)
__global__ void bn_apply_kernel(const float* __restrict__ h, const float* __restrict__ scl,
                                const float* __restrict__ shf, const float* __restrict__ csrc,
                                float* __restrict__ xs, int nRows) {
    long long gid = (long long)blockIdx.x * blockDim.x + threadIdx.x;
    long long total = (long long)nRows * (NDIM / 4);
    if (gid >= total) return;
    int row = (int)(gid >> 6);
    int j4 = (int)(gid & 63);
    float4 hv = ((const float4*)h)[gid];
    float4 sv = ((const float4*)scl)[j4];
    float4 fv = ((const float4*)shf)[j4];
    float c = csrc[row];
    float4 o;
    o.x = fmaxf(fmaf(hv.x, sv.x, fv.x), 0.0f) * c;
    o.y = fmaxf(fmaf(hv.y, sv.y, fv.y), 0.0f) * c;
    o.z = fmaxf(fmaf(hv.z, sv.z, fv.z), 0.0f) * c;
    o.w = fmaxf(fmaf(hv.w, sv.w, fv.w), 0.0f) * c;
    ((float4*)xs)[gid] = o;
}

// ---------------- host orchestration ----------------

extern "C" void kernel_launch(void* const* d_in, const int* in_sizes, int n_in,
                              void* d_out, int out_size, void* d_ws, size_t ws_size,
                              hipStream_t stream) {
    const float* x   = (const float*)d_in[0];
    const int*   src = (const int*)d_in[1];
    const int*   dst = (const int*)d_in[2];
    const float* W1  = (const float*)d_in[3];
    const float* b1  = (const float*)d_in[4];
    const float* W2  = (const float*)d_in[5];
    const float* b2  = (const float*)d_in[6];
    const float* W3  = (const float*)d_in[7];
    const float* b3  = (const float*)d_in[8];
    const float* g1  = (const float*)d_in[9];
    const float* be1 = (const float*)d_in[10];
    const float* g2  = (const float*)d_in[11];
    const float* be2 = (const float*)d_in[12];

    const int nNodes = in_sizes[0] / NDIM;   // 50000
    const int nE     = in_sizes[1];          // 800000
    const long long feat = (long long)nNodes * NDIM;

    // workspace carve-up
    float* ws = (float*)d_ws;
    float* outdeg = ws;                 ws += nNodes;
    float* indeg  = ws;                 ws += nNodes;
    float* c_src  = ws;                 ws += nNodes;
    float* c_dst  = ws;                 ws += nNodes;
    float* sum    = ws;                 ws += NDIM;
    float* sumsq  = ws;                 ws += NDIM;
    float* scl    = ws;                 ws += NDIM;
    float* shf    = ws;                 ws += NDIM;
    float* xs     = ws;                 ws += feat;
    float* agg    = ws;                 ws += feat;
    float* h      = ws;                 ws += feat;

    const int nThreads = 256;
    const int rowBlocks  = (nNodes * (NDIM / 4) + nThreads - 1) / nThreads;  // 12500
    const int scatBlocks = (int)(((long long)nE * 64 + nThreads - 1) / nThreads);
    dim3 gemmGrid((nNodes + 127) / 128, NDIM / 16);
    const float invN = 1.0f / (float)nNodes;

    // degrees + norm coefficients
    zero_kernel<<<256, nThreads, 0, stream>>>(outdeg, 2LL * nNodes);  // outdeg+indeg contiguous
    deg_kernel<<<(nE + nThreads - 1) / nThreads, nThreads, 0, stream>>>(src, dst, outdeg, indeg, nE);
    cnorm_kernel<<<(nNodes + nThreads - 1) / nThreads, nThreads, 0, stream>>>(outdeg, indeg, c_src, c_dst, nNodes);

    // ---- layer 1 ----
    scale_rows_kernel<<<rowBlocks, nThreads, 0, stream>>>(x, c_src, xs, nNodes);
    zero_kernel<<<4096, nThreads, 0, stream>>>(agg, feat);
    scatter_kernel<<<scatBlocks, nThreads, 0, stream>>>(xs, src, dst, agg, nE);
    wmma_gemm_kernel<<<gemmGrid, nThreads, 0, stream>>>(agg, c_dst, W1, b1, h, nNodes);
    zero_kernel<<<2, nThreads, 0, stream>>>(sum, 2LL * NDIM);  // sum+sumsq contiguous
    bn_stats_kernel<<<512, nThreads, 0, stream>>>(h, sum, sumsq, nNodes);
    bn_finalize_kernel<<<1, nThreads, 0, stream>>>(sum, sumsq, g1, be1, scl, shf, invN);
    bn_apply_kernel<<<rowBlocks, nThreads, 0, stream>>>(h, scl, shf, c_src, xs, nNodes);

    // ---- layer 2 ----
    zero_kernel<<<4096, nThreads, 0, stream>>>(agg, feat);
    scatter_kernel<<<scatBlocks, nThreads, 0, stream>>>(xs, src, dst, agg, nE);
    wmma_gemm_kernel<<<gemmGrid, nThreads, 0, stream>>>(agg, c_dst, W2, b2, h, nNodes);
    zero_kernel<<<2, nThreads, 0, stream>>>(sum, 2LL * NDIM);
    bn_stats_kernel<<<512, nThreads, 0, stream>>>(h, sum, sumsq, nNodes);
    bn_finalize_kernel<<<1, nThreads, 0, stream>>>(sum, sumsq, g2, be2, scl, shf, invN);
    bn_apply_kernel<<<rowBlocks, nThreads, 0, stream>>>(h, scl, shf, c_src, xs, nNodes);

    // ---- layer 3 (no BN/ReLU) ----
    zero_kernel<<<4096, nThreads, 0, stream>>>(agg, feat);
    scatter_kernel<<<scatBlocks, nThreads, 0, stream>>>(xs, src, dst, agg, nE);
    wmma_gemm_kernel<<<gemmGrid, nThreads, 0, stream>>>(agg, c_dst, W3, b3, (float*)d_out, nNodes);
}